// GINE_48137993453656
// MI455X (gfx1250) — compile-verified
//
#include <hip/hip_runtime.h>

#define N_NODES 200000
#define N_EDGES 2000000
#define NLAYER  5

typedef float v2f __attribute__((ext_vector_type(2)));
typedef float v8f __attribute__((ext_vector_type(8)));

// D = A(16x4,f32) * B(4x16,f32) + C(16x16,f32)
static __device__ __forceinline__ v8f wmma4(v2f a, v2f b, v8f c) {
  return __builtin_amdgcn_wmma_f32_16x16x4_f32(false, a, false, b, (short)0, c, false, false);
}

// A fragment: row-major [rows x ld], 16 rows from row0, K-slice [k, k+4)
// lane layout: m = lane&15 ; lanes 0-15 hold K = k+{0,1}, lanes 16-31 hold K = k+{2,3}
static __device__ __forceinline__ v2f load_a(const float* M, int ld, int row0, int k, int lane) {
  int m  = lane & 15;
  int kg = (lane >> 4) << 1;
  const float* p = M + (size_t)(row0 + m) * ld + (k + kg);
  v2f a; a.x = p[0]; a.y = p[1];
  return a;
}

// B fragment: row-major weight [K x ldw], K-slice [k, k+4), 16 cols at ncol
static __device__ __forceinline__ v2f load_b(const float* W, int ldw, int k, int ncol, int lane) {
  int n  = (lane & 15) + ncol;
  int kg = (lane >> 4) << 1;
  v2f b;
  b.x = W[(size_t)(k + kg)     * ldw + n];
  b.y = W[(size_t)(k + kg + 1) * ldw + n];
  return b;
}

// ---------------- object encoder: h = (relu(relu(x@oW1+b)@oW2+b))@oW3+b ----------------
__global__ void k_obj(const float* __restrict__ x,
                      const float* __restrict__ oW1, const float* __restrict__ ob1,
                      const float* __restrict__ oW2, const float* __restrict__ ob2,
                      const float* __restrict__ oW3, const float* __restrict__ ob3,
                      float* __restrict__ h, int N) {
  __shared__ float lds[8][16 * 32];
  const int wave = threadIdx.x >> 5, lane = threadIdx.x & 31;
  const int nbase = (blockIdx.x * 8 + wave) * 16;
  const bool act = nbase < N;
  float* my = lds[wave];
  const int n0 = lane & 15;
  const int rbase = (lane >> 4) * 8;

  if (act) {
    v8f c0 = {}, c1 = {};
    #pragma unroll
    for (int k = 0; k < 12; k += 4) {
      v2f a = load_a(x, 12, nbase, k, lane);
      c0 = wmma4(a, load_b(oW1, 32, k, 0, lane), c0);
      c1 = wmma4(a, load_b(oW1, 32, k, 16, lane), c1);
    }
    float b0 = ob1[n0], b1 = ob1[n0 + 16];
    #pragma unroll
    for (int r = 0; r < 8; ++r) {
      my[(rbase + r) * 32 + n0]      = fmaxf(c0[r] + b0, 0.f);
      my[(rbase + r) * 32 + n0 + 16] = fmaxf(c1[r] + b1, 0.f);
    }
  }
  __syncthreads();
  v8f d0 = {}, d1 = {};
  if (act) {
    #pragma unroll
    for (int k = 0; k < 32; k += 4) {
      v2f a = load_a(my, 32, 0, k, lane);
      d0 = wmma4(a, load_b(oW2, 32, k, 0, lane), d0);
      d1 = wmma4(a, load_b(oW2, 32, k, 16, lane), d1);
    }
  }
  __syncthreads();
  if (act) {
    float b0 = ob2[n0], b1 = ob2[n0 + 16];
    #pragma unroll
    for (int r = 0; r < 8; ++r) {
      my[(rbase + r) * 32 + n0]      = fmaxf(d0[r] + b0, 0.f);
      my[(rbase + r) * 32 + n0 + 16] = fmaxf(d1[r] + b1, 0.f);
    }
  }
  __syncthreads();
  if (act) {
    v8f e0 = {}, e1 = {};
    #pragma unroll
    for (int k = 0; k < 32; k += 4) {
      v2f a = load_a(my, 32, 0, k, lane);
      e0 = wmma4(a, load_b(oW3, 32, k, 0, lane), e0);
      e1 = wmma4(a, load_b(oW3, 32, k, 16, lane), e1);
    }
    float b0 = ob3[n0], b1 = ob3[n0 + 16];
    #pragma unroll
    for (int r = 0; r < 8; ++r) {
      size_t row = (size_t)(nbase + rbase + r);
      h[row * 32 + n0]      = e0[r] + b0;
      h[row * 32 + n0 + 16] = e1[r] + b1;
    }
  }
}

// ------- fused edge phase: ef = relu(ea@eW1+b)@eW2+b ; msg = relu(h[src]+ef) ; agg[dst]+=msg -------
__global__ void k_edge(const float* __restrict__ ea, const int* __restrict__ ei,
                       const float* __restrict__ eW1, const float* __restrict__ eb1,
                       const float* __restrict__ eW2, const float* __restrict__ eb2,
                       const float* __restrict__ h, float* __restrict__ agg, int E) {
  __shared__ float lds[8][16 * 32];
  const int wave = threadIdx.x >> 5, lane = threadIdx.x & 31;
  const int ebase = (blockIdx.x * 8 + wave) * 16;   // E divisible by 128: no guard
  float* my = lds[wave];
  const int n0 = lane & 15;
  const int rbase = (lane >> 4) * 8;

  // edge encoder layer 1: K=4
  v8f c0 = {}, c1 = {};
  {
    v2f a = load_a(ea, 4, ebase, 0, lane);
    c0 = wmma4(a, load_b(eW1, 32, 0, 0, lane), c0);
    c1 = wmma4(a, load_b(eW1, 32, 0, 16, lane), c1);
  }
  {
    float b0 = eb1[n0], b1 = eb1[n0 + 16];
    #pragma unroll
    for (int r = 0; r < 8; ++r) {
      my[(rbase + r) * 32 + n0]      = fmaxf(c0[r] + b0, 0.f);
      my[(rbase + r) * 32 + n0 + 16] = fmaxf(c1[r] + b1, 0.f);
    }
  }
  __syncthreads();
  // edge encoder layer 2: K=32
  v8f d0 = {}, d1 = {};
  #pragma unroll
  for (int k = 0; k < 32; k += 4) {
    v2f a = load_a(my, 32, 0, k, lane);
    d0 = wmma4(a, load_b(eW2, 32, k, 0, lane), d0);
    d1 = wmma4(a, load_b(eW2, 32, k, 16, lane), d1);
  }
  // fused GINE message + scatter-add
  float b0 = eb2[n0], b1 = eb2[n0 + 16];
  #pragma unroll
  for (int r = 0; r < 8; ++r) {
    int e = ebase + rbase + r;
    int s = ei[e];
    int d = ei[E + e];
    float m0 = fmaxf(h[(size_t)s * 32 + n0]      + d0[r] + b0, 0.f);
    float m1 = fmaxf(h[(size_t)s * 32 + n0 + 16] + d1[r] + b1, 0.f);
    atomicAdd(&agg[(size_t)d * 32 + n0],      m0);
    atomicAdd(&agg[(size_t)d * 32 + n0 + 16], m1);
  }
}

// ---------------- y = ((1+eps)*h + agg) @ hW1 + hb1   [N,64] ----------------
__global__ void k_hfunc1(const float* __restrict__ h, const float* __restrict__ agg,
                         const float* __restrict__ eps, int layer,
                         const float* __restrict__ W, const float* __restrict__ b,
                         float* __restrict__ y, int N) {
  const int wave = threadIdx.x >> 5, lane = threadIdx.x & 31;
  const int nbase = (blockIdx.x * 8 + wave) * 16;
  if (nbase >= N) return;                 // wave-uniform: EXEC stays full for WMMA
  const float ep = 1.0f + eps[layer];
  v8f c[4] = {};
  #pragma unroll
  for (int k = 0; k < 32; k += 4) {
    v2f ah = load_a(h,   32, nbase, k, lane);
    v2f ag = load_a(agg, 32, nbase, k, lane);
    v2f a; a.x = fmaf(ep, ah.x, ag.x); a.y = fmaf(ep, ah.y, ag.y);
    #pragma unroll
    for (int t = 0; t < 4; ++t)
      c[t] = wmma4(a, load_b(W, 64, k, t * 16, lane), c[t]);
  }
  const int n0 = lane & 15;
  const int rbase = (lane >> 4) * 8;
  #pragma unroll
  for (int t = 0; t < 4; ++t) {
    float bb = b[n0 + t * 16];
    #pragma unroll
    for (int r = 0; r < 8; ++r)
      y[(size_t)(nbase + rbase + r) * 64 + n0 + t * 16] = c[t][r] + bb;
  }
}

// ---------------- hpre = relu( relu(y*s1+t1) @ hW2 + hb2 ) ----------------
__global__ void k_hfunc2(const float* __restrict__ y,
                         const float* __restrict__ s1, const float* __restrict__ t1,
                         const float* __restrict__ W, const float* __restrict__ b,
                         float* __restrict__ hpre, int N) {
  const int wave = threadIdx.x >> 5, lane = threadIdx.x & 31;
  const int nbase = (blockIdx.x * 8 + wave) * 16;
  if (nbase >= N) return;
  const int kg = (lane >> 4) << 1;
  v8f c0 = {}, c1 = {};
  #pragma unroll
  for (int k = 0; k < 64; k += 4) {
    float sa = s1[k + kg],     ta = t1[k + kg];
    float sb = s1[k + kg + 1], tb = t1[k + kg + 1];
    v2f a0 = load_a(y, 64, nbase, k, lane);
    v2f a; a.x = fmaxf(fmaf(a0.x, sa, ta), 0.f);
           a.y = fmaxf(fmaf(a0.y, sb, tb), 0.f);
    c0 = wmma4(a, load_b(W, 32, k, 0, lane), c0);
    c1 = wmma4(a, load_b(W, 32, k, 16, lane), c1);
  }
  const int n0 = lane & 15, rbase = (lane >> 4) * 8;
  float b0 = b[n0], b1 = b[n0 + 16];
  #pragma unroll
  for (int r = 0; r < 8; ++r) {
    size_t row = (size_t)(nbase + rbase + r);
    hpre[row * 32 + n0]      = fmaxf(c0[r] + b0, 0.f);
    hpre[row * 32 + n0 + 16] = fmaxf(c1[r] + b1, 0.f);
  }
}

// ---------------- per-column sum / sumsq over [N,C] ----------------
__global__ void k_colstats(const float* __restrict__ x, int N, int C,
                           float* __restrict__ sum, float* __restrict__ sq) {
  __shared__ float ls[256], lq[256];
  const int tid = threadIdx.x;
  const int c = tid % C;
  const int sub = tid / C;
  const int subcnt = blockDim.x / C;
  float s = 0.f, q = 0.f;
  for (int r = blockIdx.x * subcnt + sub; r < N; r += gridDim.x * subcnt) {
    float v = x[(size_t)r * C + c];
    s += v; q += v * v;
  }
  ls[tid] = s; lq[tid] = q;
  __syncthreads();
  for (int st = subcnt >> 1; st > 0; st >>= 1) {
    if (sub < st) { ls[tid] += ls[tid + st * C]; lq[tid] += lq[tid + st * C]; }
    __syncthreads();
  }
  if (sub == 0) { atomicAdd(&sum[c], ls[tid]); atomicAdd(&sq[c], lq[tid]); }
}

// ---------------- fold BN params: s = g*rsqrt(var+eps), t = b - mean*s ----------------
__global__ void k_bnparams(const float* __restrict__ sum, const float* __restrict__ sq,
                           const float* __restrict__ g, const float* __restrict__ b,
                           float* __restrict__ s, float* __restrict__ t, int C, float invN) {
  int c = threadIdx.x;
  if (c >= C) return;
  float m  = sum[c] * invN;
  float v  = sq[c] * invN - m * m;
  float sc = g[c] * rsqrtf(v + 1e-5f);
  s[c] = sc;
  t[c] = b[c] - m * sc;
}

// ---------------- h = hpre * s + t  (per-column affine) ----------------
__global__ void k_bnapply(const float* __restrict__ xin, const float* __restrict__ s,
                          const float* __restrict__ t, float* __restrict__ xout, int n4) {
  int i = blockIdx.x * blockDim.x + threadIdx.x;
  if (i >= n4) return;
  float4 v = ((const float4*)xin)[i];
  int c = (i << 2) & 31;
  v.x = fmaf(v.x, s[c],     t[c]);
  v.y = fmaf(v.y, s[c + 1], t[c + 1]);
  v.z = fmaf(v.z, s[c + 2], t[c + 2]);
  v.w = fmaf(v.w, s[c + 3], t[c + 3]);
  ((float4*)xout)[i] = v;
}

// ---------------- out = relu(h@f1W+f1b) @ f2W + f2b   [N,6] ----------------
__global__ void k_final(const float* __restrict__ h,
                        const float* __restrict__ f1W, const float* __restrict__ f1b,
                        const float* __restrict__ f2W, const float* __restrict__ f2b,
                        float* __restrict__ out, int N) {
  __shared__ float lds[8][16 * 32];
  const int wave = threadIdx.x >> 5, lane = threadIdx.x & 31;
  const int nbase = (blockIdx.x * 8 + wave) * 16;
  const bool act = nbase < N;
  float* my = lds[wave];
  const int n0 = lane & 15;
  const int rbase = (lane >> 4) * 8;
  const int kg = (lane >> 4) << 1;

  if (act) {
    v8f c0 = {}, c1 = {};
    #pragma unroll
    for (int k = 0; k < 32; k += 4) {
      v2f a = load_a(h, 32, nbase, k, lane);
      c0 = wmma4(a, load_b(f1W, 32, k, 0, lane), c0);
      c1 = wmma4(a, load_b(f1W, 32, k, 16, lane), c1);
    }
    float b0 = f1b[n0], b1 = f1b[n0 + 16];
    #pragma unroll
    for (int r = 0; r < 8; ++r) {
      my[(rbase + r) * 32 + n0]      = fmaxf(c0[r] + b0, 0.f);
      my[(rbase + r) * 32 + n0 + 16] = fmaxf(c1[r] + b1, 0.f);
    }
  }
  __syncthreads();
  if (act) {
    v8f c = {};
    #pragma unroll
    for (int k = 0; k < 32; k += 4) {
      v2f a = load_a(my, 32, 0, k, lane);
      v2f b;   // f2W is [32 x 6]: pad columns 6..15 with zeros
      b.x = (n0 < 6) ? f2W[(k + kg) * 6 + n0]     : 0.f;
      b.y = (n0 < 6) ? f2W[(k + kg + 1) * 6 + n0] : 0.f;
      c = wmma4(a, b, c);
    }
    if (n0 < 6) {
      float bb = f2b[n0];
      #pragma unroll
      for (int r = 0; r < 8; ++r)
        out[(size_t)(nbase + rbase + r) * 6 + n0] = c[r] + bb;
    }
  }
}

__global__ void k_zero(float* __restrict__ p, int n) {
  int i = blockIdx.x * blockDim.x + threadIdx.x;
  if (i < n) p[i] = 0.f;
}

extern "C" void kernel_launch(void* const* d_in, const int* in_sizes, int n_in,
                              void* d_out, int out_size, void* d_ws, size_t ws_size,
                              hipStream_t stream) {
  const int N = N_NODES, E = N_EDGES;
  const float* x   = (const float*)d_in[0];
  const int*   ei  = (const int*)  d_in[1];
  const float* ea  = (const float*)d_in[2];
  const float* oW1 = (const float*)d_in[3];
  const float* ob1 = (const float*)d_in[4];
  const float* oW2 = (const float*)d_in[5];
  const float* ob2 = (const float*)d_in[6];
  const float* oW3 = (const float*)d_in[7];
  const float* ob3 = (const float*)d_in[8];
  const float* eW1 = (const float*)d_in[9];
  const float* eb1 = (const float*)d_in[10];
  const float* eW2 = (const float*)d_in[11];
  const float* eb2 = (const float*)d_in[12];
  const float* eps = (const float*)d_in[13];
  const float* hW1 = (const float*)d_in[14];
  const float* hb1 = (const float*)d_in[15];
  const float* hg  = (const float*)d_in[16];
  const float* hbb = (const float*)d_in[17];
  const float* hW2 = (const float*)d_in[18];
  const float* hb2 = (const float*)d_in[19];
  const float* bng = (const float*)d_in[20];
  const float* bnb = (const float*)d_in[21];
  const float* f1W = (const float*)d_in[22];
  const float* f1b = (const float*)d_in[23];
  const float* f2W = (const float*)d_in[24];
  const float* f2b = (const float*)d_in[25];

  float* ws   = (float*)d_ws;
  float* h    = ws;                        // N*32
  float* agg  = h    + (size_t)N * 32;     // N*32
  float* y    = agg  + (size_t)N * 32;     // N*64
  float* hpre = y    + (size_t)N * 64;     // N*32
  float* st   = hpre + (size_t)N * 32;     // stats scratch
  float *sum1 = st,       *sq1 = st + 64,  *s1 = st + 128, *t1 = st + 192;
  float *sum2 = st + 256, *sq2 = st + 288, *s2 = st + 320, *t2 = st + 352;

  dim3 blk(256);
  const int nodeBlocks = (N / 16 + 7) / 8;   // 1563 blocks of 8 waves x 16 rows
  const int edgeBlocks = E / 128;            // 15625, exact
  const float invN = 1.0f / (float)N;

  k_obj<<<nodeBlocks, blk, 0, stream>>>(x, oW1, ob1, oW2, ob2, oW3, ob3, h, N);

  for (int i = 0; i < NLAYER; ++i) {
    k_zero<<<(N * 32 + 255) / 256, blk, 0, stream>>>(agg, N * 32);
    k_edge<<<edgeBlocks, blk, 0, stream>>>(ea, ei, eW1 + i * 4 * 32, eb1 + i * 32,
                                           eW2 + i * 32 * 32, eb2 + i * 32, h, agg, E);
    k_zero<<<1, blk, 0, stream>>>(sum1, 128);
    k_hfunc1<<<nodeBlocks, blk, 0, stream>>>(h, agg, eps, i,
                                             hW1 + i * 32 * 64, hb1 + i * 64, y, N);
    k_colstats<<<512, blk, 0, stream>>>(y, N, 64, sum1, sq1);
    k_bnparams<<<1, dim3(64), 0, stream>>>(sum1, sq1, hg + i * 64, hbb + i * 64, s1, t1, 64, invN);
    k_zero<<<1, blk, 0, stream>>>(sum2, 64);
    k_hfunc2<<<nodeBlocks, blk, 0, stream>>>(y, s1, t1, hW2 + i * 64 * 32, hb2 + i * 32, hpre, N);
    k_colstats<<<512, blk, 0, stream>>>(hpre, N, 32, sum2, sq2);
    k_bnparams<<<1, dim3(64), 0, stream>>>(sum2, sq2, bng + i * 32, bnb + i * 32, s2, t2, 32, invN);
    k_bnapply<<<(N * 32 / 4 + 255) / 256, blk, 0, stream>>>(hpre, s2, t2, h, N * 32 / 4);
  }

  k_final<<<nodeBlocks, blk, 0, stream>>>(h, f1W, f1b, f2W, f2b, (float*)d_out, N);
}